// MambaBlock_30932354465938
// MI455X (gfx1250) — compile-verified
//
#include <hip/hip_runtime.h>
#include <hip/hip_bf16.h>
#include <math.h>

// ---------------------------------------------------------------------------
// Mamba block for MI455X (gfx1250, wave32, WMMA, async global->LDS).
// Pipeline:
//   1) xz = x @ W_in.T                      (WMMA f32 GEMM, 2048x4096x1024)
//   2) depthwise conv(K=4, causal) + SiLU   (elementwise)
//   3) xp = xh @ W_x.T                      (WMMA f32 GEMM, N=40 padded)
//   4) delta = softplus(dt_in @ W_dt.T + b) (tiny K=8)
//   5) chunked selective scan (3 phases)    (65536-way parallel)
//   6) out = y @ W_out.T                    (WMMA f32 GEMM, 2048x1024x2048)
// ---------------------------------------------------------------------------

typedef __attribute__((ext_vector_type(2))) float v2f;
typedef __attribute__((ext_vector_type(8))) float v8f;

#define WMMA_BM 128
#define WMMA_BN 128
#define WMMA_BK 32
#define LDS_STRIDE 36   // floats; 16B-aligned rows for async b128, bank-friendly

__device__ __forceinline__ v8f wmma_f32_16x16x4(v2f a, v2f b, v8f c) {
#if defined(__HIP_DEVICE_COMPILE__)
  // 8 args: (neg_a, A, neg_b, B, c_mod, C, reuse_a, reuse_b)
  return __builtin_amdgcn_wmma_f32_16x16x4_f32(false, a, false, b, (short)0, c,
                                               false, false);
#else
  (void)a; (void)b;
  return c;
#endif
}

// Async 16-byte global -> LDS copy (CDNA5 TDM-adjacent path, ASYNCcnt).
// LDS generic pointers carry the LDS byte offset in their low 32 bits.
__device__ __forceinline__ void async_copy16(const float* __restrict__ gsrc,
                                             float* __restrict__ ldst) {
#if defined(__gfx1250__)
  const unsigned lds_off = (unsigned)(size_t)ldst;
  asm volatile("global_load_async_to_lds_b128 %0, %1, off"
               :
               : "v"(lds_off), "v"(gsrc)
               : "memory");
#else
  *reinterpret_cast<float4*>(ldst) = *reinterpret_cast<const float4*>(gsrc);
#endif
}

__device__ __forceinline__ void wait_async0() {
#if defined(__gfx1250__)
  asm volatile("s_wait_asynccnt 0x0" ::: "memory");
#endif
}

__device__ __forceinline__ float dev_sigmoid(float x) {
  return 1.0f / (1.0f + expf(-x));
}

// ---------------------------------------------------------------------------
// C[M x N] = A[M x K] @ B[N x K]^T        (A,B row-major, K contiguous)
// Block tile 128x128x32, 256 threads = 8 waves in 4(M) x 2(N) grid,
// each wave computes 32x64 via 2x4 WMMA 16x16 accumulators.
// Double-buffered LDS; tile k+1 streams in via async-LDS during WMMA on k.
// M,K must be multiples of 128/32; N may be ragged (guarded loads/stores).
// ---------------------------------------------------------------------------
__global__ __launch_bounds__(256)
void gemm_nt_wmma(const float* __restrict__ A, const float* __restrict__ B,
                  float* __restrict__ C, int M, int N, int K, int ldc)
{
  __shared__ float As[2][WMMA_BM * LDS_STRIDE];
  __shared__ float Bs[2][WMMA_BN * LDS_STRIDE];

  const int tid  = threadIdx.x;
  const int lane = tid & 31;
  const int wave = tid >> 5;
  const int wm   = wave >> 1;   // 0..3  (M direction)
  const int wn   = wave & 1;    // 0..1  (N direction)
  const int l16  = lane & 15;
  const int lh   = lane >> 4;   // half-wave select

  const int bm = blockIdx.y * WMMA_BM;
  const int bn = blockIdx.x * WMMA_BN;

  // global->LDS mapping: 256 threads, each moves one float4 per 32-row pass
  const int lr = tid >> 3;        // 0..31 rows per pass
  const int lc = (tid & 7) * 4;   // column offset 0..28

  // Ragged N: zero the never-loaded B rows once (both buffers).
  if (bn + WMMA_BN > N) {
#pragma unroll
    for (int p = 0; p < 4; ++p) {
      const int r = p * 32 + lr;
      if (bn + r >= N) {
        float* d0 = &Bs[0][r * LDS_STRIDE + lc];
        float* d1 = &Bs[1][r * LDS_STRIDE + lc];
        d0[0] = d0[1] = d0[2] = d0[3] = 0.0f;
        d1[0] = d1[1] = d1[2] = d1[3] = 0.0f;
      }
    }
  }

  auto issue_tiles = [&](int buf, int k0) {
#pragma unroll
    for (int p = 0; p < 4; ++p) {
      const int r = p * 32 + lr;
      async_copy16(&A[(size_t)(bm + r) * K + (k0 + lc)],
                   &As[buf][r * LDS_STRIDE + lc]);
    }
#pragma unroll
    for (int p = 0; p < 4; ++p) {
      const int r = p * 32 + lr;
      if (bn + r < N)
        async_copy16(&B[(size_t)(bn + r) * K + (k0 + lc)],
                     &Bs[buf][r * LDS_STRIDE + lc]);
    }
  };

  v8f acc[2][4];
#pragma unroll
  for (int i = 0; i < 2; ++i)
#pragma unroll
    for (int j = 0; j < 4; ++j)
      acc[i][j] = (v8f)0.0f;

  issue_tiles(0, 0);
  int cur = 0;

  for (int k0 = 0; k0 < K; k0 += WMMA_BK) {
    // Publish tile `cur` (and retire all reads of the other buffer).
    wait_async0();
    __syncthreads();

    if (k0 + WMMA_BK < K)
      issue_tiles(cur ^ 1, k0 + WMMA_BK);   // overlaps with WMMA below

    const float* as = As[cur];
    const float* bs = Bs[cur];

#pragma unroll
    for (int kk = 0; kk < WMMA_BK; kk += 4) {
      // A frag (16x4 f32): lanes 0-15 hold K={kk,kk+1}, lanes 16-31 K={kk+2,kk+3}
      v2f afrag[2], bfrag[4];
#pragma unroll
      for (int i = 0; i < 2; ++i) {
        const int m = wm * 32 + i * 16 + l16;
        afrag[i] = *reinterpret_cast<const v2f*>(
            &as[m * LDS_STRIDE + kk + 2 * lh]);
      }
#pragma unroll
      for (int j = 0; j < 4; ++j) {
        const int n = wn * 64 + j * 16 + l16;
        bfrag[j] = *reinterpret_cast<const v2f*>(
            &bs[n * LDS_STRIDE + kk + 2 * lh]);
      }
#pragma unroll
      for (int i = 0; i < 2; ++i)
#pragma unroll
        for (int j = 0; j < 4; ++j)
          acc[i][j] = wmma_f32_16x16x4(afrag[i], bfrag[j], acc[i][j]);
    }
    cur ^= 1;
  }

  // D layout: VGPR e -> M = e (lanes 0-15) / e+8 (lanes 16-31), N = lane&15
#pragma unroll
  for (int i = 0; i < 2; ++i) {
#pragma unroll
    for (int j = 0; j < 4; ++j) {
      const int col = bn + wn * 64 + j * 16 + l16;
      if (col < N) {
#pragma unroll
        for (int e = 0; e < 8; ++e) {
          const int row = bm + wm * 32 + i * 16 + 8 * lh + e;
          C[(size_t)row * ldc + col] = acc[i][j][e];
        }
      }
    }
  }
}

// ---------------------------------------------------------------------------
// Depthwise causal conv (K=4) + bias + SiLU.  xz holds [xh | z] rows of 4096.
// idx -> (b*1024+t)*2048 + d
// ---------------------------------------------------------------------------
__global__ __launch_bounds__(256)
void conv_silu_k(const float* __restrict__ xz, const float* __restrict__ cw,
                 const float* __restrict__ cb, float* __restrict__ xh)
{
  const int idx = blockIdx.x * 256 + threadIdx.x;   // 0 .. 4194303
  const int d  = idx & 2047;
  const int bt = idx >> 11;       // b*1024 + t
  const int t  = bt & 1023;

  float acc = cb[d];
#pragma unroll
  for (int j = 0; j < 4; ++j) {
    const int tt = t - 3 + j;
    if (tt >= 0)
      acc += cw[d * 4 + j] * xz[(size_t)(bt - 3 + j) * 4096 + d];
  }
  xh[idx] = acc * dev_sigmoid(acc);
}

// ---------------------------------------------------------------------------
// delta = softplus(dt_in @ W_dt.T + b_dt);  dt_in = xp[..., 0:8]
// ---------------------------------------------------------------------------
__global__ __launch_bounds__(256)
void delta_k(const float* __restrict__ xp, const float* __restrict__ W_dt,
             const float* __restrict__ b_dt, float* __restrict__ delta)
{
  const int idx = blockIdx.x * 256 + threadIdx.x;   // 0 .. 4194303
  const int d  = idx & 2047;
  const int bt = idx >> 11;

  const float* r = &xp[(size_t)bt * 40];
  const float* w = &W_dt[d * 8];
  float s = b_dt[d];
#pragma unroll
  for (int j = 0; j < 8; ++j) s += r[j] * w[j];
  delta[idx] = (s > 20.0f) ? s : log1pf(expf(s));
}

// ---------------------------------------------------------------------------
// Chunked selective scan: 16 chunks of 64 timesteps.
// Phase A: local scan from h=0 per chunk; emit chunk decay P = exp(a_n*sum d)
//          and chunk-final local state.
// lane -> (chunk, b, d); 65536 lanes total.
// ---------------------------------------------------------------------------
__global__ __launch_bounds__(256)
void scan_phaseA(const float* __restrict__ delta, const float* __restrict__ xh,
                 const float* __restrict__ xp, const float* __restrict__ A_log,
                 float* __restrict__ Pbuf, float* __restrict__ Hbuf)
{
  const int gid   = blockIdx.x * 256 + threadIdx.x;  // 0 .. 65535
  const int ch    = gid & 4095;                      // b*2048 + d
  const int chunk = gid >> 12;
  const int b     = ch >> 11;
  const int d     = ch & 2047;

  float a[16];
#pragma unroll
  for (int n = 0; n < 16; ++n) a[n] = -expf(A_log[n]);

  float h[16];
#pragma unroll
  for (int n = 0; n < 16; ++n) h[n] = 0.0f;
  float sd = 0.0f;

  const int t0 = chunk * 64;
  for (int t = t0; t < t0 + 64; ++t) {
    const size_t row = (size_t)(b * 1024 + t);
    const float dv = delta[row * 2048 + d];
    const float xv = xh[row * 2048 + d];
    const float* xpr = &xp[row * 40];
    const float dx = dv * xv;
    sd += dv;
#pragma unroll
    for (int n = 0; n < 16; ++n) {
      const float dA = expf(dv * a[n]);
      h[n] = dA * h[n] + dx * xpr[8 + n];
    }
  }

  const size_t o = ((size_t)chunk * 4096 + ch) * 16;
#pragma unroll
  for (int n = 0; n < 16; ++n) {
    Pbuf[o + n] = expf(a[n] * sd);
    Hbuf[o + n] = h[n];
  }
}

// ---------------------------------------------------------------------------
// Combine: sequential over 16 chunks, parallel over (channel, state).
// Seed[c] = state entering chunk c.
// ---------------------------------------------------------------------------
__global__ __launch_bounds__(256)
void scan_combine(const float* __restrict__ Pbuf, const float* __restrict__ Hbuf,
                  float* __restrict__ Seed)
{
  const int gid = blockIdx.x * 256 + threadIdx.x;   // 0 .. 65535 = ch*16+n
  float H = 0.0f;
#pragma unroll
  for (int c = 0; c < 16; ++c) {
    const size_t idx = (size_t)c * 65536 + gid;
    Seed[idx] = H;
    H = Pbuf[idx] * H + Hbuf[idx];
  }
}

// ---------------------------------------------------------------------------
// Phase C: replay each chunk from its seed; fuse y = (h.C + D*x) * silu(z).
// ---------------------------------------------------------------------------
__global__ __launch_bounds__(256)
void scan_phaseC(const float* __restrict__ delta, const float* __restrict__ xh,
                 const float* __restrict__ xp, const float* __restrict__ A_log,
                 const float* __restrict__ Dp, const float* __restrict__ xz,
                 const float* __restrict__ Seed, float* __restrict__ Y)
{
  const int gid   = blockIdx.x * 256 + threadIdx.x;  // 0 .. 65535
  const int ch    = gid & 4095;
  const int chunk = gid >> 12;
  const int b     = ch >> 11;
  const int d     = ch & 2047;

  float a[16];
#pragma unroll
  for (int n = 0; n < 16; ++n) a[n] = -expf(A_log[n]);

  float h[16];
  const size_t so = ((size_t)chunk * 4096 + ch) * 16;
#pragma unroll
  for (int n = 0; n < 16; ++n) h[n] = Seed[so + n];

  const float Dd = Dp[d];

  const int t0 = chunk * 64;
  for (int t = t0; t < t0 + 64; ++t) {
    const size_t row = (size_t)(b * 1024 + t);
    const float dv = delta[row * 2048 + d];
    const float xv = xh[row * 2048 + d];
    const float* xpr = &xp[row * 40];
    const float dx = dv * xv;
    float acc = 0.0f;
#pragma unroll
    for (int n = 0; n < 16; ++n) {
      const float dA = expf(dv * a[n]);
      h[n] = dA * h[n] + dx * xpr[8 + n];
      acc += h[n] * xpr[24 + n];
    }
    float y = acc + Dd * xv;
    const float zv = xz[row * 4096 + 2048 + d];
    y *= zv * dev_sigmoid(zv);
    Y[row * 2048 + d] = y;
  }
}

// ---------------------------------------------------------------------------
// Workspace layout (floats)
// ---------------------------------------------------------------------------
static constexpr size_t OFF_XZ    = 0;                        // 2048*4096
static constexpr size_t OFF_XH    = OFF_XZ    + 8388608;      // 2048*2048
static constexpr size_t OFF_DELTA = OFF_XH    + 4194304;      // 2048*2048
static constexpr size_t OFF_XP    = OFF_DELTA + 4194304;      // 2048*40 (pad)
static constexpr size_t OFF_P     = OFF_XP    + 131072;       // 16*4096*16
static constexpr size_t OFF_HF    = OFF_P     + 1048576;
static constexpr size_t OFF_SEED  = OFF_HF    + 1048576;
static constexpr size_t OFF_Y     = OFF_SEED  + 1048576;      // 2048*2048

extern "C" void kernel_launch(void* const* d_in, const int* in_sizes, int n_in,
                              void* d_out, int out_size, void* d_ws, size_t ws_size,
                              hipStream_t stream) {
  (void)in_sizes; (void)n_in; (void)out_size; (void)ws_size;

  const float* x      = (const float*)d_in[0];
  const float* W_in   = (const float*)d_in[1];
  const float* conv_w = (const float*)d_in[2];
  const float* conv_b = (const float*)d_in[3];
  const float* W_x    = (const float*)d_in[4];
  const float* W_dt   = (const float*)d_in[5];
  const float* b_dt   = (const float*)d_in[6];
  const float* A_log  = (const float*)d_in[7];
  const float* Dp     = (const float*)d_in[8];
  const float* W_out  = (const float*)d_in[9];

  float* out = (float*)d_out;
  float* w   = (float*)d_ws;

  float* xz    = w + OFF_XZ;
  float* xh    = w + OFF_XH;
  float* delta = w + OFF_DELTA;
  float* xp    = w + OFF_XP;
  float* Pb    = w + OFF_P;
  float* Hb    = w + OFF_HF;
  float* Seed  = w + OFF_SEED;
  float* Y     = w + OFF_Y;

  // 1) xz = x @ W_in.T   (M=2048, N=4096, K=1024)
  gemm_nt_wmma<<<dim3(32, 16), 256, 0, stream>>>(x, W_in, xz, 2048, 4096, 1024, 4096);
  // 2) depthwise conv + SiLU
  conv_silu_k<<<16384, 256, 0, stream>>>(xz, conv_w, conv_b, xh);
  // 3) xp = xh @ W_x.T   (M=2048, N=40, K=2048), ldc=40
  gemm_nt_wmma<<<dim3(1, 16), 256, 0, stream>>>(xh, W_x, xp, 2048, 40, 2048, 40);
  // 4) delta = softplus(dt_in @ W_dt.T + b_dt)
  delta_k<<<16384, 256, 0, stream>>>(xp, W_dt, b_dt, delta);
  // 5) chunked scan
  scan_phaseA<<<256, 256, 0, stream>>>(delta, xh, xp, A_log, Pb, Hb);
  scan_combine<<<256, 256, 0, stream>>>(Pb, Hb, Seed);
  scan_phaseC<<<256, 256, 0, stream>>>(delta, xh, xp, A_log, Dp, xz, Seed, Y);
  // 6) out = y @ W_out.T (M=2048, N=1024, K=2048)
  gemm_nt_wmma<<<dim3(8, 16), 256, 0, stream>>>(Y, W_out, out, 2048, 1024, 2048, 1024);
}